// PytorchBasedGMM_30562987278340
// MI455X (gfx1250) — compile-verified
//
#include <hip/hip_runtime.h>
#include <hip/hip_bf16.h>
#include <math.h>

typedef __attribute__((ext_vector_type(2))) float v2f;
typedef __attribute__((ext_vector_type(8))) float v8f;

#define NPTS    1000000
#define FDIM    48            // 36 tri + 8 linear + 1 const + 3 pad
#define NCHUNK  (NPTS / 16)   // 62500, N divisible by 16
#define NBLOCKS 1024
#define WS_PART_OFF 4096      // float offset of block partials in workspace

__host__ __device__ constexpr int tri_idx(int d, int e) {  // d <= e
    return d * 8 - (d * (d - 1)) / 2 + (e - d);
}

// ---------------------------------------------------------------------------
// Phase 1: per-component preprocessing (tiny): cov build, 8x8 inverse + det,
// softmax of prior, pack coefficient matrix W[48][64] into workspace.
// ---------------------------------------------------------------------------
__global__ void gmm_setup(const float* __restrict__ prior_logits,
                          const float* __restrict__ mus,
                          const float* __restrict__ cov_raw,
                          float* __restrict__ W) {
    __shared__ float red[64];
    __shared__ float GA[64 * 64];
    __shared__ float GI[64 * 64];
    const int k = threadIdx.x;  // 64 threads, one component each

    // log-softmax of prior_logits via LDS tree (deterministic)
    float logit = prior_logits[k];
    red[k] = logit;
    __syncthreads();
    for (int s = 32; s > 0; s >>= 1) {
        if (k < s) red[k] = fmaxf(red[k], red[k + s]);
        __syncthreads();
    }
    float mx = red[0];
    __syncthreads();
    red[k] = __expf(logit - mx);
    __syncthreads();
    for (int s = 32; s > 0; s >>= 1) {
        if (k < s) red[k] += red[k + s];
        __syncthreads();
    }
    float logp = logit - (mx + __logf(red[0]));

    // covariance_k = exp(C) on diag + tanh((C + C^T)/2) off diag
    float* A = &GA[k * 64];
    float* I = &GI[k * 64];
    for (int i = 0; i < 8; ++i)
        for (int j = 0; j < 8; ++j) {
            float cij = cov_raw[k * 64 + i * 8 + j];
            float cji = cov_raw[k * 64 + j * 8 + i];
            A[i * 8 + j] = (i == j) ? __expf(cij) : tanhf(0.5f * (cij + cji));
            I[i * 8 + j] = (i == j) ? 1.0f : 0.0f;
        }

    // Gauss-Jordan inverse with determinant (matrix is SPD, no pivoting)
    float det = 1.0f;
    for (int i = 0; i < 8; ++i) {
        float piv = A[i * 8 + i];
        det *= piv;
        float ip = 1.0f / piv;
        for (int j = 0; j < 8; ++j) { A[i * 8 + j] *= ip; I[i * 8 + j] *= ip; }
        for (int r = 0; r < 8; ++r) {
            if (r == i) continue;
            float f = A[r * 8 + i];
            for (int j = 0; j < 8; ++j) {
                A[r * 8 + j] -= f * A[i * 8 + j];
                I[r * 8 + j] -= f * I[i * 8 + j];
            }
        }
    }

    float mu[8];
    #pragma unroll
    for (int d = 0; d < 8; ++d) mu[d] = mus[k * 8 + d];
    float Sm[8];
    float cst = 0.0f;
    #pragma unroll
    for (int d = 0; d < 8; ++d) {
        float s = 0.0f;
        #pragma unroll
        for (int e = 0; e < 8; ++e) s += I[d * 8 + e] * mu[e];
        Sm[d] = s;
        cst += mu[d] * s;
    }

    // W[f][k]: t[n,k] = phi(x_n) . W[:,k] = log(prior_k * norm_k) - 0.5*quad
    #pragma unroll
    for (int d = 0; d < 8; ++d)
        #pragma unroll
        for (int e = d; e < 8; ++e) {
            float c = (d == e) ? -0.5f : -1.0f;  // symmetry: off-diag counted once
            W[tri_idx(d, e) * 64 + k] = c * I[d * 8 + e];
        }
    #pragma unroll
    for (int d = 0; d < 8; ++d) W[(36 + d) * 64 + k] = Sm[d];
    const float LOG2PI = 1.8378770664093453f;
    W[44 * 64 + k] = logp - 4.0f * LOG2PI - 0.5f * __logf(det) - 0.5f * cst;
    W[45 * 64 + k] = 0.0f;
    W[46 * 64 + k] = 0.0f;
    W[47 * 64 + k] = 0.0f;
}

// ---------------------------------------------------------------------------
// Phase 2: main GEMM [N,64] = phi[N,48] x W[48,64] via V_WMMA_F32_16X16X4_F32,
// fused with per-row logsumexp over the 64 components.
// ---------------------------------------------------------------------------
__global__ void __launch_bounds__(256, 1)
gmm_main(const float* __restrict__ X,
         const float* __restrict__ Wg,
         float* __restrict__ partials) {
    __shared__ float sW[FDIM * 64];        // 12 KB: coefficient matrix
    __shared__ float sPhi[8][16 * FDIM];   // 24 KB: per-wave phi staging
    __shared__ float sWaveSum[8];

    const int tid = threadIdx.x;
    for (int i = tid; i < FDIM * 64; i += 256) sW[i] = Wg[i];
    __syncthreads();

    const int lane  = tid & 31;
    const int wslot = tid >> 5;
    const int p = lane & 15;   // row (A) / column (B,C,D) within 16
    const int h = lane >> 4;   // half-wave: K-offset for A/B, row-offset for C/D

    // Preload all B fragments (4 component tiles x 12 K-steps), kept in VGPRs.
    // 32-bit B 4x16 layout: V0 holds K=2h, V1 holds K=2h+1; lane p = column.
    v2f B[4][12];
    #pragma unroll
    for (int t = 0; t < 4; ++t)
        #pragma unroll
        for (int s = 0; s < 12; ++s) {
            B[t][s].x = sW[(4 * s + 2 * h + 0) * 64 + 16 * t + p];
            B[t][s].y = sW[(4 * s + 2 * h + 1) * 64 + 16 * t + p];
        }

    float* phiBase = &sPhi[wslot][0];
    const long gwave  = (long)blockIdx.x * 8 + wslot;
    const long nwaves = (long)gridDim.x * 8;

    float acc_total = 0.0f;

    for (long chunk = gwave; chunk < NCHUNK; chunk += nwaves) {
        const float* xp = X + chunk * 128;  // 16 points x 8 dims

        // Lanes 0..15 each build phi for one point into LDS.
        if (h == 0) {
            const float4* xv4 = (const float4*)(xp + p * 8);
            float4 a0 = xv4[0], a1 = xv4[1];
            float x[8] = {a0.x, a0.y, a0.z, a0.w, a1.x, a1.y, a1.z, a1.w};
            float* pr = phiBase + p * FDIM;
            #pragma unroll
            for (int d = 0; d < 8; ++d)
                #pragma unroll
                for (int e = d; e < 8; ++e)
                    pr[tri_idx(d, e)] = x[d] * x[e];
            #pragma unroll
            for (int d = 0; d < 8; ++d) pr[36 + d] = x[d];
            pr[44] = 1.0f; pr[45] = 0.0f; pr[46] = 0.0f; pr[47] = 0.0f;
        }
        // DS ops are in-order within a wave; just stop the compiler reordering.
        __builtin_amdgcn_wave_barrier();
        asm volatile("" ::: "memory");

        // Prefetch next chunk of X while we do WMMA work.
        long nxt = chunk + nwaves;
        if (nxt < NCHUNK) __builtin_prefetch(X + nxt * 128 + lane * 4, 0, 0);

        v8f acc[4];
        #pragma unroll
        for (int t = 0; t < 4; ++t)
            #pragma unroll
            for (int j = 0; j < 8; ++j) acc[t][j] = 0.0f;

        // A 16x4 f32 layout: lane row m = lane%16; V0:K=2h, V1:K=2h+1.
        const float* pr = phiBase + p * FDIM;
        #pragma unroll
        for (int s = 0; s < 12; ++s) {
            v2f a;
            a.x = pr[4 * s + 2 * h + 0];
            a.y = pr[4 * s + 2 * h + 1];
            #pragma unroll
            for (int t = 0; t < 4; ++t)
                acc[t] = __builtin_amdgcn_wmma_f32_16x16x4_f32(
                    false, a, false, B[t][s], (short)0, acc[t], false, false);
        }

        // logsumexp over 64 components per point.
        // C/D layout: lane holds row (j + 8h), column (16t + p) -> a full row
        // lives in the 16 lanes of one half-wave across the 4 tiles.
        float rows = 0.0f;
        #pragma unroll
        for (int j = 0; j < 8; ++j) {
            float m = fmaxf(fmaxf(acc[0][j], acc[1][j]),
                            fmaxf(acc[2][j], acc[3][j]));
            #pragma unroll
            for (int off = 1; off <= 8; off <<= 1)
                m = fmaxf(m, __shfl_xor(m, off, 32));
            float sum = __expf(acc[0][j] - m) + __expf(acc[1][j] - m) +
                        __expf(acc[2][j] - m) + __expf(acc[3][j] - m);
            #pragma unroll
            for (int off = 1; off <= 8; off <<= 1)
                sum += __shfl_xor(sum, off, 32);
            rows += m + __logf(sum);
        }
        float tot = rows + __shfl_xor(rows, 16, 32);  // combine both halves
        acc_total += tot;  // identical in all lanes
    }

    if (lane == 0) sWaveSum[wslot] = acc_total;
    __syncthreads();
    if (tid == 0) {
        float s = 0.0f;
        #pragma unroll
        for (int w = 0; w < 8; ++w) s += sWaveSum[w];
        partials[blockIdx.x] = s;
    }
}

// ---------------------------------------------------------------------------
// Phase 3: deterministic fixed-order final reduction.
// ---------------------------------------------------------------------------
__global__ void gmm_finalize(const float* __restrict__ partials,
                             float* __restrict__ out) {
    __shared__ float red[256];
    int t = threadIdx.x;
    float s = 0.0f;
    for (int i = t; i < NBLOCKS; i += 256) s += partials[i];
    red[t] = s;
    __syncthreads();
    for (int st = 128; st > 0; st >>= 1) {
        if (t < st) red[t] += red[t + st];
        __syncthreads();
    }
    if (t == 0) out[0] = red[0];
}

extern "C" void kernel_launch(void* const* d_in, const int* in_sizes, int n_in,
                              void* d_out, int out_size, void* d_ws, size_t ws_size,
                              hipStream_t stream) {
    const float* X     = (const float*)d_in[0];  // [N,8]
    const float* prior = (const float*)d_in[1];  // [64]
    const float* mus   = (const float*)d_in[2];  // [64,8]
    const float* cov   = (const float*)d_in[3];  // [64,8,8]
    float* ws       = (float*)d_ws;
    float* W        = ws;                  // 48*64 floats
    float* partials = ws + WS_PART_OFF;    // NBLOCKS floats

    gmm_setup<<<1, 64, 0, stream>>>(prior, mus, cov, W);
    gmm_main<<<NBLOCKS, 256, 0, stream>>>(X, W, partials);
    gmm_finalize<<<1, 256, 0, stream>>>(partials, (float*)d_out);
}